// MambaForecaster_50646254354921
// MI455X (gfx1250) — compile-verified
//
#include <hip/hip_runtime.h>
#include <hip/hip_bf16.h>
#include <math.h>

// ---------------------------------------------------------------------------
// Mamba forward for MI455X (gfx1250, wave32).
//  * GEMMs: V_WMMA_F32_16X16X4_F32, operands staged through LDS (coalesced,
//    zero-filled for N-tail -> branch-free inner loop of ds_load + wmma).
//  * Selective scan: chunked 3-pass linear-diagonal scan (exp(A*sum_delta)
//    chunk transition) -> 512 blocks instead of 16 on the serial dimension.
//  * global_prefetch_b8 via __builtin_prefetch on the streamed scan operands.
// ---------------------------------------------------------------------------

typedef __attribute__((ext_vector_type(2))) float v2f;
typedef __attribute__((ext_vector_type(8))) float v8f;

#define BATCH   16
#define LSEQ    2048
#define TOKENS  (BATCH * LSEQ)   // 32768
#define DM      64
#define ED      128
#define NST     16
#define RNK     4
#define NLAYERS 4
#define DBC_W   36               // R + 2N
#define EPSQ    1e-5f

#define CCH     32               // scan chunks per sequence
#define LC      (LSEQ / CCH)     // 64 steps per chunk

#define LDS_PAD 4                // floats of row padding (bank-conflict free)
#define KMAX    128

__device__ __forceinline__ float silu_f(float v)     { return v / (1.f + __expf(-v)); }
__device__ __forceinline__ float softplus_f(float v) { return (v > 20.f) ? v : log1pf(__expf(v)); }

// ---------------------------------------------------------------------------
// f32 WMMA GEMM:  O[m,n] = sum_k A[m,k] * W[n,k]   (W row-major [N,K])
// Block = 128 threads (4 waves). Block tile: 16 (M) x 64 (N). K <= 128.
// A tile (16xK) and W tile (64xK) staged in LDS, coalesced, W zero-filled
// past N so the K-loop has no guards.
// EPI: 0=none  1=+bias  2=softplus(.+bias)  3=in-place residual (O += result)
// ---------------------------------------------------------------------------
template <int EPI>
__global__ __launch_bounds__(128)
void gemm_f32_wmma(const float* __restrict__ A, const float* __restrict__ W,
                   const float* __restrict__ bias, float* __restrict__ O,
                   int N, int K, int lda, int ldw, int ldo)
{
    __shared__ float sA[16 * (KMAX + LDS_PAD)];
    __shared__ float sW[64 * (KMAX + LDS_PAD)];

    const int tid  = threadIdx.x;
    const int lane = tid & 31;
    const int wave = tid >> 5;
    const int nblk = blockIdx.x * 64;
    const int tile_m = blockIdx.y * 16;
    const int KP = K + LDS_PAD;

    // Stage A tile (16 x K), coalesced across the block.
    for (int i = tid; i < 16 * K; i += 128) {
        int r = i / K, c = i - r * K;
        sA[r * KP + c] = A[(size_t)(tile_m + r) * lda + c];
    }
    // Stage W tile (64 x K), zero-filled for rows >= N.
    for (int i = tid; i < 64 * K; i += 128) {
        int r = i / K, c = i - r * K;
        int gn = nblk + r;
        sW[r * KP + c] = (gn < N) ? W[(size_t)gn * ldw + c] : 0.f;
    }
    __syncthreads();

    // Fragment indices (CDNA5 ISA 7.12.2, 32-bit A 16x4 / C-D 16x16 layouts).
    const int mrow = lane & 15;
    const int koff = (lane >> 4) << 1;      // lanes 0-15 -> K{0,1}; 16-31 -> K{2,3}
    const float* __restrict__ aRow = sA + mrow * KP + koff;
    const float* __restrict__ wRow = sW + (wave * 16 + mrow) * KP + koff;

    v8f acc = {};
    #pragma unroll 4
    for (int k = 0; k < K; k += 4) {
        v2f a, b;
        a.x = aRow[k];  a.y = aRow[k + 1];
        b.x = wRow[k];  b.y = wRow[k + 1];
        acc = __builtin_amdgcn_wmma_f32_16x16x4_f32(false, a, false, b,
                                                    (short)0, acc, false, false);
    }

    const int nst = nblk + wave * 16 + (lane & 15);
    if (nst < N) {
        const int mbase = tile_m + ((lane >> 4) << 3);
        #pragma unroll
        for (int v = 0; v < 8; ++v) {
            size_t oi = (size_t)(mbase + v) * ldo + nst;
            float val = acc[v];
            if (EPI == 1) val += bias[nst];
            if (EPI == 2) val = softplus_f(val + bias[nst]);
            if (EPI == 3) val += O[oi];
            O[oi] = val;
        }
    }
}

// ---------------------------------------------------------------------------
// RMSNorm: one thread per token (Dm = 64).
// ---------------------------------------------------------------------------
__global__ void rmsnorm_kernel(const float* __restrict__ x, const float* __restrict__ nw,
                               float* __restrict__ xn)
{
    int t = blockIdx.x * blockDim.x + threadIdx.x;
    if (t >= TOKENS) return;
    const float* xp = x + (size_t)t * DM;
    float ss = 0.f;
    #pragma unroll 8
    for (int i = 0; i < DM; ++i) ss += xp[i] * xp[i];
    float r = rsqrtf(ss * (1.f / DM) + EPSQ);
    float* op = xn + (size_t)t * DM;
    #pragma unroll 8
    for (int i = 0; i < DM; ++i) op[i] = xp[i] * r * nw[i];
}

// ---------------------------------------------------------------------------
// Depthwise causal conv (DC=4) + SiLU.  xi half of xz = columns [0,128).
// ---------------------------------------------------------------------------
__global__ void conv_silu_kernel(const float* __restrict__ xz, const float* __restrict__ cw,
                                 const float* __restrict__ cb, float* __restrict__ xi)
{
    int idx = blockIdx.x * blockDim.x + threadIdx.x;     // over TOKENS*ED
    if (idx >= TOKENS * ED) return;
    int e = idx & (ED - 1);
    int t = idx >> 7;
    int l = t & (LSEQ - 1);
    float acc = cb[e];
    #pragma unroll
    for (int j = 0; j < 4; ++j) {
        int ll = l - 3 + j;
        if (ll >= 0) acc += xz[(size_t)(t - 3 + j) * (2 * ED) + e] * cw[e * 4 + j];
    }
    xi[idx] = silu_f(acc);
}

// ---------------------------------------------------------------------------
// Chunked selective scan, pass 1: per-chunk scan from h=0.
// Produces S[b,c,e,n] (chunk-final state) and sumd[b,c,e] (sum of delta).
// Block = 128 threads = one (batch, chunk); thread-per-channel e.
// ---------------------------------------------------------------------------
__global__ __launch_bounds__(ED)
void scan_partial(const float* __restrict__ dbc, const float* __restrict__ delta,
                  const float* __restrict__ xi, const float* __restrict__ A_log,
                  float* __restrict__ S, float* __restrict__ sumd)
{
    __shared__ float sB[NST];
    const int b = blockIdx.x, c = blockIdx.y, e = threadIdx.x;

    float Ae[NST], h[NST];
    #pragma unroll
    for (int n = 0; n < NST; ++n) { Ae[n] = -__expf(A_log[e * NST + n]); h[n] = 0.f; }
    float sd = 0.f;

    for (int i = 0; i < LC; ++i) {
        size_t tok = (size_t)b * LSEQ + c * LC + i;
        if (threadIdx.x < NST) sB[threadIdx.x] = dbc[tok * DBC_W + RNK + threadIdx.x];
        __syncthreads();
        float d  = delta[tok * ED + e];
        float xv = xi[tok * ED + e];
        __builtin_prefetch(&delta[(tok + 1) * ED + e], 0, 0);
        __builtin_prefetch(&xi[(tok + 1) * ED + e], 0, 0);
        sd += d;
        float dx = d * xv;
        #pragma unroll
        for (int n = 0; n < NST; ++n)
            h[n] = __expf(d * Ae[n]) * h[n] + dx * sB[n];
        __syncthreads();
    }

    size_t base = (size_t)(b * CCH + c) * ED + e;
    #pragma unroll
    for (int n = 0; n < NST; ++n) S[base * NST + n] = h[n];
    sumd[base] = sd;
}

// ---------------------------------------------------------------------------
// Pass 2: chunk-serial prefix (32 steps). H0[b,c,e,n] = state entering chunk c.
// Chunk transition product = exp(A[n] * sum_delta).
// ---------------------------------------------------------------------------
__global__ __launch_bounds__(ED)
void scan_prefix(const float* __restrict__ S, const float* __restrict__ sumd,
                 const float* __restrict__ A_log, float* __restrict__ H0)
{
    const int b = blockIdx.x, e = threadIdx.x;
    float Ae[NST], h[NST];
    #pragma unroll
    for (int n = 0; n < NST; ++n) { Ae[n] = -__expf(A_log[e * NST + n]); h[n] = 0.f; }

    for (int c = 0; c < CCH; ++c) {
        size_t base = (size_t)(b * CCH + c) * ED + e;
        float sd = sumd[base];
        #pragma unroll
        for (int n = 0; n < NST; ++n) {
            H0[base * NST + n] = h[n];
            h[n] = __expf(sd * Ae[n]) * h[n] + S[base * NST + n];
        }
    }
}

// ---------------------------------------------------------------------------
// Pass 3: replay each chunk from its true h0, emit y = h.C + Dp*xi.
// ---------------------------------------------------------------------------
__global__ __launch_bounds__(ED)
void scan_emit(const float* __restrict__ dbc, const float* __restrict__ delta,
               const float* __restrict__ xi, const float* __restrict__ A_log,
               const float* __restrict__ Dp, const float* __restrict__ H0,
               float* __restrict__ y)
{
    __shared__ float sB[NST];
    __shared__ float sC[NST];
    const int b = blockIdx.x, c = blockIdx.y, e = threadIdx.x;

    float Ae[NST], h[NST];
    size_t base = (size_t)(b * CCH + c) * ED + e;
    #pragma unroll
    for (int n = 0; n < NST; ++n) {
        Ae[n] = -__expf(A_log[e * NST + n]);
        h[n]  = H0[base * NST + n];
    }
    const float dpe = Dp[e];

    for (int i = 0; i < LC; ++i) {
        size_t tok = (size_t)b * LSEQ + c * LC + i;
        if (threadIdx.x < 2 * NST) {
            float v = dbc[tok * DBC_W + RNK + threadIdx.x];
            if (threadIdx.x < NST) sB[threadIdx.x] = v;
            else                   sC[threadIdx.x - NST] = v;
        }
        __syncthreads();
        float d  = delta[tok * ED + e];
        float xv = xi[tok * ED + e];
        __builtin_prefetch(&delta[(tok + 1) * ED + e], 0, 0);
        __builtin_prefetch(&xi[(tok + 1) * ED + e], 0, 0);
        float dx = d * xv;
        float yacc = 0.f;
        #pragma unroll
        for (int n = 0; n < NST; ++n) {
            h[n] = __expf(d * Ae[n]) * h[n] + dx * sB[n];
            yacc += h[n] * sC[n];
        }
        y[tok * ED + e] = yacc + dpe * xv;
        __syncthreads();
    }
}

// ---------------------------------------------------------------------------
// Gating: y *= silu(z), z = xz columns [128,256).
// ---------------------------------------------------------------------------
__global__ void gate_kernel(float* __restrict__ y, const float* __restrict__ xz)
{
    int idx = blockIdx.x * blockDim.x + threadIdx.x;
    if (idx >= TOKENS * ED) return;
    int e = idx & (ED - 1);
    int t = idx >> 7;
    float z = xz[(size_t)t * (2 * ED) + ED + e];
    y[idx] *= silu_f(z);
}

__global__ void copy_kernel(const float* __restrict__ src, float* __restrict__ dst, int n)
{
    int i = blockIdx.x * blockDim.x + threadIdx.x;
    if (i < n) dst[i] = src[i];
}

__global__ void gather_last_kernel(const float* __restrict__ x, float* __restrict__ last)
{
    int i = blockIdx.x * blockDim.x + threadIdx.x;   // over BATCH*DM
    if (i >= BATCH * DM) return;
    int b = i / DM, d = i % DM;
    last[i] = x[((size_t)b * LSEQ + (LSEQ - 1)) * DM + d];
}

// ---------------------------------------------------------------------------
// kernel_launch
// ---------------------------------------------------------------------------
extern "C" void kernel_launch(void* const* d_in, const int* in_sizes, int n_in,
                              void* d_out, int out_size, void* d_ws, size_t ws_size,
                              hipStream_t stream)
{
    (void)in_sizes; (void)n_in; (void)out_size; (void)ws_size;

    const float* x_in   = (const float*)d_in[0];
    const float* norm_w = (const float*)d_in[1];
    const float* in_w   = (const float*)d_in[2];
    const float* conv_w = (const float*)d_in[3];
    const float* conv_b = (const float*)d_in[4];
    const float* xp_w   = (const float*)d_in[5];
    const float* dt_w   = (const float*)d_in[6];
    const float* dt_b   = (const float*)d_in[7];
    const float* A_log  = (const float*)d_in[8];
    const float* Dp     = (const float*)d_in[9];
    const float* out_w  = (const float*)d_in[10];
    const float* fc_w   = (const float*)d_in[11];
    const float* fc_b   = (const float*)d_in[12];
    float* out = (float*)d_out;

    // Workspace layout (floats); total ~28.5M floats (~114 MB).
    float* ws    = (float*)d_ws;
    float* x_res = ws;                                   // TOKENS*DM
    float* xn    = x_res + (size_t)TOKENS * DM;          // TOKENS*DM
    float* xz    = xn    + (size_t)TOKENS * DM;          // TOKENS*2*ED
    float* xi    = xz    + (size_t)TOKENS * 2 * ED;      // TOKENS*ED
    float* dbc   = xi    + (size_t)TOKENS * ED;          // TOKENS*DBC_W
    float* delta = dbc   + (size_t)TOKENS * DBC_W;       // TOKENS*ED
    float* yb    = delta + (size_t)TOKENS * ED;          // TOKENS*ED
    float* Sbuf  = yb    + (size_t)TOKENS * ED;          // BATCH*CCH*ED*NST
    float* H0buf = Sbuf  + (size_t)BATCH * CCH * ED * NST;
    float* sdbuf = H0buf + (size_t)BATCH * CCH * ED * NST; // BATCH*CCH*ED
    float* last  = sdbuf + (size_t)BATCH * CCH * ED;       // BATCH*DM

    const int ew = 256;   // elementwise block size

    copy_kernel<<<(TOKENS * DM + ew - 1) / ew, ew, 0, stream>>>(x_in, x_res, TOKENS * DM);

    for (int l = 0; l < NLAYERS; ++l) {
        // 1) RMSNorm
        rmsnorm_kernel<<<TOKENS / 128, 128, 0, stream>>>(x_res, norm_w + (size_t)l * DM, xn);

        // 2) in_proj: [TOKENS,64] @ [256,64]^T -> xz [TOKENS,256]
        gemm_f32_wmma<0><<<dim3(256 / 64, TOKENS / 16), 128, 0, stream>>>(
            xn, in_w + (size_t)l * 2 * ED * DM, nullptr, xz,
            2 * ED, DM, DM, DM, 2 * ED);

        // 3) depthwise causal conv + SiLU -> xi [TOKENS,128]
        conv_silu_kernel<<<(TOKENS * ED + ew - 1) / ew, ew, 0, stream>>>(
            xz, conv_w + (size_t)l * ED * 4, conv_b + (size_t)l * ED, xi);

        // 4) x_proj: [TOKENS,128] @ [36,128]^T -> dbc [TOKENS,36] (N-tail zero-filled)
        gemm_f32_wmma<0><<<dim3(1, TOKENS / 16), 128, 0, stream>>>(
            xi, xp_w + (size_t)l * DBC_W * ED, nullptr, dbc,
            DBC_W, ED, ED, ED, DBC_W);

        // 5) dt_proj + softplus: [TOKENS,4] @ [128,4]^T + b -> delta [TOKENS,128]
        gemm_f32_wmma<2><<<dim3(ED / 64, TOKENS / 16), 128, 0, stream>>>(
            dbc, dt_w + (size_t)l * ED * RNK, dt_b + (size_t)l * ED, delta,
            ED, RNK, DBC_W, RNK, ED);

        // 6) chunked selective scan -> yb [TOKENS,128]
        const float* Al = A_log + (size_t)l * ED * NST;
        scan_partial<<<dim3(BATCH, CCH), ED, 0, stream>>>(dbc, delta, xi, Al, Sbuf, sdbuf);
        scan_prefix<<<BATCH, ED, 0, stream>>>(Sbuf, sdbuf, Al, H0buf);
        scan_emit<<<dim3(BATCH, CCH), ED, 0, stream>>>(
            dbc, delta, xi, Al, Dp + (size_t)l * ED, H0buf, yb);

        // 7) gate: yb *= silu(z)
        gate_kernel<<<(TOKENS * ED + ew - 1) / ew, ew, 0, stream>>>(yb, xz);

        // 8) out_proj + residual (in place on x_res): x += yb @ [64,128]^T
        gemm_f32_wmma<3><<<dim3(1, TOKENS / 16), 128, 0, stream>>>(
            yb, out_w + (size_t)l * DM * ED, nullptr, x_res,
            DM, ED, ED, ED, DM);
    }

    // Final head: last token per batch, then FC + bias -> out [16,64]
    gather_last_kernel<<<(BATCH * DM + ew - 1) / ew, ew, 0, stream>>>(x_res, last);
    gemm_f32_wmma<1><<<dim3(1, 1), 128, 0, stream>>>(
        last, fc_w, fc_b, out, DM, DM, DM, DM, DM);
}